// LIFLayer_52243982189246
// MI455X (gfx1250) — compile-verified
//
#include <hip/hip_runtime.h>

#define NN 4096
#define TT 300
#define MTILES 19              // ceil(300/16)
#define NTILES 256             // 4096/16
#define WAVES_PER_BLOCK 8
#define NB 16                  // persistent blocks (one neuron per thread, 16*256=4096)
#define TPB 256

typedef float v2f __attribute__((ext_vector_type(2)));
typedef float v8f __attribute__((ext_vector_type(8)));

// ---------------------------------------------------------------------------
// Phase 1: X[t][n] = sum_k inp[t][k] * w[k][n], batched over all 300 timesteps
// with exact-f32 WMMA (V_WMMA_F32_16X16X4_F32). Layouts per CDNA5 ISA 7.12.2:
//   A 16x4 f32 : lanes 0-15 hold M=lane, K={0,1} in v0,v1; lanes 16-31 K={2,3}
//   B 4x16 f32 : row-striped across lanes (lane = N), K split across VGPR/halves
//   C 16x16 f32: VGPR r -> M=r (lanes 0-15), M=r+8 (lanes 16-31), N=lane&15
// Out-of-range M rows (300..303 in the last tile) read clamped row 0: the
// garbage lands only in C rows that the store loop discards, so no per-
// iteration masking is needed and the inner loop is loads + cvt + wmma only.
// ---------------------------------------------------------------------------
__global__ __launch_bounds__(256) void gemm_inp_w(const int* __restrict__ inp,
                                                  const float* __restrict__ w,
                                                  float* __restrict__ X) {
  const int lane = threadIdx.x & 31;
  const int wave = threadIdx.x >> 5;
  const int mtile = blockIdx.y;
  const int ntile = blockIdx.x * WAVES_PER_BLOCK + wave;
  const int m0 = mtile * 16;
  const int n0 = ntile * 16;

  const int mrow = m0 + (lane & 15);
  const int arow = (mrow < TT) ? mrow : 0;  // clamp address only
  const int koff = (lane >> 4) * 2;         // 0 or 2
  const int ncol = n0 + (lane & 15);

  const int* __restrict__ aptr = inp + (size_t)arow * NN + koff;
  const float* __restrict__ bptr = w + (size_t)koff * NN + ncol;

  v8f c = {};
#pragma unroll 8
  for (int k0 = 0; k0 < NN; k0 += 4) {
    v2f a;
    a.x = (float)aptr[k0];
    a.y = (float)aptr[k0 + 1];
    v2f b;
    b.x = bptr[(size_t)k0 * NN];
    b.y = bptr[(size_t)(k0 + 1) * NN];
    // 8 args: (neg_a, A, neg_b, B, c_mod, C, reuse_a, reuse_b)
    c = __builtin_amdgcn_wmma_f32_16x16x4_f32(false, a, false, b, (short)0, c,
                                              false, false);
  }

#pragma unroll
  for (int r = 0; r < 8; ++r) {
    int m = m0 + r + ((lane >> 4) << 3);
    if (m < TT) X[(size_t)m * NN + ncol] = c[r];
  }
}

// ---------------------------------------------------------------------------
// Device-wide sense-reversing barrier (all NB blocks are resident on MI455X).
// ---------------------------------------------------------------------------
__device__ inline void grid_barrier(unsigned* cnt, unsigned* gen) {
  __syncthreads();
  if (threadIdx.x == 0) {
    __threadfence();                       // release my stores
    unsigned g = atomicAdd(gen, 0u);       // read generation BEFORE arriving
    unsigned prev = atomicAdd(cnt, 1u);
    if (prev == NB - 1) {
      atomicExch(cnt, 0u);
      __threadfence();
      atomicAdd(gen, 1u);                  // release everyone
    } else {
      while (atomicAdd(gen, 0u) == g) __builtin_amdgcn_s_sleep(1);
    }
    __threadfence();                       // acquire others' stores
  }
  __syncthreads();
}

// ---------------------------------------------------------------------------
// Phase 2: persistent LIF recurrence. 4096 threads, one neuron each; v/rt/t in
// registers for all 300 steps. Per step: x = X[t][j] + sum over ACTIVE rows of
// w_rec (spikes are binary -> masked column sum, coalesced 16KB row reads from
// the 192MB L2 where both weight matrices stay resident); the sum is skipped
// entirely when this neuron is refractory since x would be zeroed anyway.
// Active list rebuilt each step with deterministic ballot compaction.
// ---------------------------------------------------------------------------
__global__ __launch_bounds__(TPB) void lif_persistent(
    const float* __restrict__ w_rec, const float* __restrict__ X,
    float* __restrict__ out_s, float* __restrict__ out_t,
    float* __restrict__ out_v, unsigned* __restrict__ ctrl,
    int* __restrict__ list) {
  const int j = blockIdx.x * TPB + threadIdx.x;  // neuron id 0..4095
  const int lane = threadIdx.x & 31;
  const int wave = threadIdx.x >> 5;

  unsigned* bar_cnt = ctrl;        // [0],[1]
  unsigned* bar_gen = ctrl + 2;    // [2],[3]
  unsigned* blk_cnt = ctrl + 16;   // 16 per-block spike counts (byte 64..127)

  __shared__ unsigned s_wave_cnt[TPB / 32];
  __shared__ int s_counts[NB];

  float v = 0.0f;          // RESET
  float rt = 0.0f;
  float trec = 1.0f;       // t initialized to ones
  const float decay = 0.99004983374916805f;  // exp(-1/100)

  for (int t = 0; t < TT; ++t) {
    // input drive is produced by the (stream-ordered) GEMM, not by the list:
    // issue its load before the barrier to hide latency under the spin-wait
    float acc = X[(size_t)t * NN + j];

    // spike list produced at end of previous step is now visible everywhere
    grid_barrier(bar_cnt + 0, bar_gen + 0);

    // stage the 16 uniform per-block counts through LDS once per step
    if (threadIdx.x < NB) s_counts[threadIdx.x] = (int)blk_cnt[threadIdx.x];
    __syncthreads();

    if (rt > 0.0f) {
      acc = 0.0f;                          // refractory gating (zeroes all of x)
    } else {
#pragma unroll 1
      for (int b = 0; b < NB; ++b) {
        const int c = s_counts[b];         // uniform across grid
        const int* lp = list + b * TPB;
        int k = 0;
        for (; k + 4 <= c; k += 4) {       // ILP: 4 independent row reads
          int i0 = lp[k], i1 = lp[k + 1], i2 = lp[k + 2], i3 = lp[k + 3];
          float r0 = w_rec[(size_t)i0 * NN + j];
          float r1 = w_rec[(size_t)i1 * NN + j];
          float r2 = w_rec[(size_t)i2 * NN + j];
          float r3 = w_rec[(size_t)i3 * NN + j];
          acc += r0; acc += r1; acc += r2; acc += r3;
        }
        for (; k < c; ++k) acc += w_rec[(size_t)lp[k] * NN + j];
      }
    }

    rt -= 1.0f;
    v = decay * v + acc;
    bool spike = (v >= 10.0f);
    float s = spike ? 1.0f : 0.0f;
    if (spike) {
      trec = (float)t * (1.0f / 300.0f);   // ts*DT/TIME
      rt = 1.0f;                           // REFRAC
      v = 0.0f;                            // RESET
    }
    size_t o = (size_t)t * NN + j;
    out_s[o] = s;
    out_t[o] = trec;
    out_v[o] = v;

    // everyone is done reading the old list; safe to overwrite it
    grid_barrier(bar_cnt + 1, bar_gen + 1);

    // deterministic intra-block compaction of spiking neuron indices
    unsigned mask = __builtin_amdgcn_ballot_w32(spike);
    if (lane == 0) s_wave_cnt[wave] = (unsigned)__popc(mask);
    __syncthreads();
    unsigned woff = 0;
#pragma unroll
    for (int ww = 0; ww < TPB / 32; ++ww)
      if (ww < wave) woff += s_wave_cnt[ww];
    unsigned prefix = (unsigned)__popc(mask & ((1u << lane) - 1u));
    if (spike) list[blockIdx.x * TPB + woff + prefix] = j;
    if (threadIdx.x == 0) {
      unsigned tot = 0;
#pragma unroll
      for (int ww = 0; ww < TPB / 32; ++ww) tot += s_wave_cnt[ww];
      blk_cnt[blockIdx.x] = tot;
    }
    __syncthreads();   // s_wave_cnt/s_counts reused next iteration
  }
}

// ---------------------------------------------------------------------------
extern "C" void kernel_launch(void* const* d_in, const int* in_sizes, int n_in,
                              void* d_out, int out_size, void* d_ws,
                              size_t ws_size, hipStream_t stream) {
  const int* inp = (const int*)d_in[0];      // [300,4096] binary int32
  const float* w = (const float*)d_in[1];    // [4096,4096] f32
  const float* w_rec = (const float*)d_in[2];// [4096,4096] f32

  float* out = (float*)d_out;
  float* out_s = out;
  float* out_t = out + (size_t)TT * NN;
  float* out_v = out + 2 * (size_t)TT * NN;

  char* ws = (char*)d_ws;
  unsigned* ctrl = (unsigned*)ws;            // barriers + per-block spike counts
  int* list = (int*)(ws + 256);              // 4096 compacted spike indices
  float* X = (float*)(ws + 32768);           // [300,4096] input-drive, 4.9 MB

  // zero barrier/count state every call (deterministic across graph replays)
  hipMemsetAsync(d_ws, 0, 256, stream);

  dim3 gemm_grid(NTILES / WAVES_PER_BLOCK, MTILES, 1);  // 32 x 19 blocks
  gemm_inp_w<<<gemm_grid, 256, 0, stream>>>(inp, w, X);

  lif_persistent<<<NB, TPB, 0, stream>>>(w_rec, X, out_s, out_t, out_v, ctrl,
                                         list);
}